// GatedLinearAttention_61040075210837
// MI455X (gfx1250) — compile-verified
//
#include <hip/hip_runtime.h>

// ---------------------------------------------------------------------------
// Gated linear attention for MI455X (gfx1250, wave32, WMMA, TDM).
// f32 throughout (reference is f32 with exp/log gating -> keep exact f32 MACs
// via V_WMMA_F32_16X16X4_F32). Whole working set (~40MB) is L2-resident.
// ---------------------------------------------------------------------------

typedef __attribute__((ext_vector_type(2))) float v2f;
typedef __attribute__((ext_vector_type(8))) float v8f;
typedef __attribute__((ext_vector_type(4))) unsigned int u32x4;
typedef __attribute__((ext_vector_type(8))) int i32x8;
typedef __attribute__((ext_vector_type(4))) int i32x4;

#define L_SEQ  1024
#define DMODEL 1024
#define NHEAD  16
#define DHEAD  64
#define CHUNK  64
#define ALD    68  // padded LDS pitch (floats): 16B-aligned, bank-conflict free

#if __has_builtin(__builtin_amdgcn_tensor_load_to_lds)
#define USE_TDM 1
#else
#define USE_TDM 0
#endif

// D = A(16x4) * B(4x16) + C, full f32.
__device__ __forceinline__ v8f wmma_f32_k4(v2f a, v2f b, v8f c) {
  return __builtin_amdgcn_wmma_f32_16x16x4_f32(false, a, false, b, (short)0, c,
                                               false, false);
}

// A fragment: A is MxK row-major, 16x4 tile at (row0,k0).
// lanes 0-15 -> M=0..15 / K={k0,k0+1}; lanes 16-31 -> M=0..15 / K={k0+2,k0+3}.
__device__ __forceinline__ v2f frag_a(const float* M, int ld, int row0, int k0,
                                      int lane) {
  const int m = lane & 15, h = lane >> 4;
  const float* p = M + (row0 + m) * ld + (k0 + 2 * h);
  v2f r; r.x = p[0]; r.y = p[1];
  return r;
}
// B fragment: B is KxN row-major, 4x16 tile at (k0,col0).
__device__ __forceinline__ v2f frag_b(const float* M, int ld, int k0, int col0,
                                      int lane) {
  const int n = lane & 15, h = lane >> 4;
  const float* p = M + (k0 + 2 * h) * ld + (col0 + n);
  v2f r; r.x = p[0]; r.y = p[ld];
  return r;
}
// B fragment where the logical B is Source^T: B[k][n] = S[n][k], S row-major.
__device__ __forceinline__ v2f frag_bt(const float* S, int ld, int k0, int col0,
                                       int lane) {
  const int n = lane & 15, h = lane >> 4;
  const float* p = S + (col0 + n) * ld + (k0 + 2 * h);
  v2f r; r.x = p[0]; r.y = p[1];
  return r;
}

#if USE_TDM
// Issue one TDM 2D tile load: 64x64 f32 tile from a row-major matrix with row
// stride DMODEL floats into LDS at byte offset lds_off, LDS row pitch = ALD
// floats (pad_enable: +4 DWORDs after every 64 DWORDs stored).
__device__ __forceinline__ void tdm_load_tile(const float* gsrc,
                                              unsigned lds_off) {
  const unsigned long long ga = (unsigned long long)gsrc;
  u32x4 g0;
  g0[0] = 1u;                                  // count=1 (valid user D#)
  g0[1] = lds_off;                             // LDS byte address
  g0[2] = (unsigned)ga;                        // global_addr[31:0]
  g0[3] = (unsigned)(ga >> 32) | (2u << 30);   // global_addr[56:32] | type=2
  i32x8 g1;
  // data_size=4B (2<<16) | pad_enable (1<<20) | pad_interval=64dw (5<<22)
  // | pad_amount=4dw (3<<25)
  g1[0] = (int)((2u << 16) | (1u << 20) | (5u << 22) | (3u << 25));
  g1[1] = (int)((DMODEL & 0xffff) << 16);      // tensor_dim0 lo16 @ [31:16]
  g1[2] = (int)((DMODEL & 0xffff) << 16);      // dim0 hi16 (=0) | dim1 lo16
  g1[3] = (int)(64u << 16);                    // dim1 hi16 (=0) | tile_dim0=64
  g1[4] = 64;                                  // tile_dim1=64, tile_dim2=0
  g1[5] = DMODEL;                              // tensor_dim0_stride lo32
  g1[6] = 0;                                   // stride0 hi | stride1 lo
  g1[7] = 0;
  const i32x4 z4 = {0, 0, 0, 0};
  const i32x8 z8 = {0, 0, 0, 0, 0, 0, 0, 0};
  __builtin_amdgcn_tensor_load_to_lds(g0, g1, z4, z4, z8, 0);
}
#endif

// ---------------------------------------------------------------------------
// Y (1024x1024) = X (1024x1024) @ W^T   (W row-major 1024x1024)
// Block = 8 waves, 64x64 output tile, LDS-staged 64-wide K blocks
// (coalesced float4 global reads, padded-pitch LDS fragments).
// ---------------------------------------------------------------------------
__global__ __launch_bounds__(256) void gemm_xwT_kernel(
    const float* __restrict__ X, const float* __restrict__ W,
    float* __restrict__ Y) {
  __shared__ float sX[64 * ALD];
  __shared__ float sW[64 * ALD];  // row-major W tile: sW[n][k]
  const int lane = threadIdx.x & 31;
  const int wave = threadIdx.x >> 5;  // 0..7
  const int bm = blockIdx.y * 64, bn = blockIdx.x * 64;
  const int m0 = (wave >> 1) * 16;
  const int n0 = (wave & 1) * 32;
  const int tr = threadIdx.x >> 2;         // staging row 0..63
  const int tc = (threadIdx.x & 3) * 16;   // staging col start

  v8f acc0 = {};
  v8f acc1 = {};
  for (int kb = 0; kb < DMODEL; kb += 64) {
    __syncthreads();  // previous K-block fully consumed
    const float4* xg = (const float4*)(X + (bm + tr) * DMODEL + kb + tc);
    const float4* wg = (const float4*)(W + (bn + tr) * DMODEL + kb + tc);
    for (int q = 0; q < 4; ++q) {
      *(float4*)&sX[tr * ALD + tc + 4 * q] = xg[q];
      *(float4*)&sW[tr * ALD + tc + 4 * q] = wg[q];
    }
    __syncthreads();
    for (int k0 = 0; k0 < 64; k0 += 4) {
      v2f a  = frag_a(sX, ALD, m0, k0, lane);
      v2f b0 = frag_bt(sW, ALD, k0, n0, lane);
      v2f b1 = frag_bt(sW, ALD, k0, n0 + 16, lane);
      acc0 = wmma_f32_k4(a, b0, acc0);
      acc1 = wmma_f32_k4(a, b1, acc1);
    }
  }
  const int n = lane & 15, half = lane >> 4;
  for (int v = 0; v < 8; ++v) {
    const int row = bm + m0 + v + 8 * half;
    Y[row * DMODEL + bn + n0 + n]      = acc0[v];
    Y[row * DMODEL + bn + n0 + 16 + n] = acc1[v];
  }
}

// ---------------------------------------------------------------------------
// Per (l,h): logg = log(clip(sigmoid(x_head . gw), 1e-6)), Rv = silu(K_head . sw)
// ---------------------------------------------------------------------------
__global__ __launch_bounds__(256) void gate_kernel(
    const float* __restrict__ x, const float* __restrict__ Kp,
    const float* __restrict__ gw, const float* __restrict__ sw,
    float* __restrict__ logg, float* __restrict__ Rv) {
  const int idx = blockIdx.x * blockDim.x + threadIdx.x;  // 0 .. L*H-1
  if (idx >= L_SEQ * NHEAD) return;
  const int l = idx >> 4, h = idx & (NHEAD - 1);
  const float* xr = x + l * DMODEL + h * DHEAD;
  const float* kr = Kp + l * DMODEL + h * DHEAD;
  float gd = 0.f, sd = 0.f;
  for (int j = 0; j < DHEAD; ++j) {
    gd += xr[j] * gw[j];
    sd += kr[j] * sw[j];
  }
  float sig = 1.f / (1.f + expf(-gd));
  sig = fmaxf(sig, 1e-6f);
  logg[idx] = logf(sig);
  Rv[idx] = sd / (1.f + expf(-sd));  // silu
}

// ---------------------------------------------------------------------------
// Per head: inclusive prefix-sum of logg over L (wave32 parallel scan),
// Gc = exp(clip(cum,-30,30)) + 1e-6 ; rr = Rv / Gc.
// ---------------------------------------------------------------------------
__global__ __launch_bounds__(32) void scan_kernel(
    const float* __restrict__ logg, const float* __restrict__ Rv,
    float* __restrict__ Gc, float* __restrict__ rr) {
  const int h = blockIdx.x;
  const int lane = threadIdx.x;
  const int base = lane * 32;
  float vals[32];
  float s = 0.f;
  for (int i = 0; i < 32; ++i) {
    vals[i] = logg[(base + i) * NHEAD + h];
    s += vals[i];
  }
  float e = s;
  for (int d = 1; d < 32; d <<= 1) {
    float t = __shfl_up(e, d, 32);
    if (lane >= d) e += t;
  }
  float run = e - s;  // exclusive prefix for this lane
  for (int i = 0; i < 32; ++i) {
    run += vals[i];
    const float lc = fminf(fmaxf(run, -30.f), 30.f);
    const float g = expf(lc) + 1e-6f;
    const int idx = (base + i) * NHEAD + h;
    Gc[idx] = g;
    rr[idx] = Rv[idx] / g;
  }
}

// ---------------------------------------------------------------------------
// Chunked GLA scan. One workgroup per head (16 waves = 4x4 grid of 16x16
// tiles over a 64x64 chunk). LDS state U[d,e] = sum_{s<=t} r_s K_s V_s^T.
// Per chunk:  O = Gc_t * ( Q@U + (tril(Q@K^T)*r) @ V ),   U += (K*r)^T @ V
// Chunk staging via the Tensor Data Mover (async, padded-pitch LDS tiles).
// ---------------------------------------------------------------------------
__global__ __launch_bounds__(512) void gla_attn_kernel(
    const float* __restrict__ Qg, const float* __restrict__ Kg,
    const float* __restrict__ Vg, const float* __restrict__ Gc,
    const float* __restrict__ rr, float* __restrict__ Og) {
  extern __shared__ float smem[];
  float* sQ = smem;                // 64 x ALD
  float* sK = smem + 4352;         // 64*68 = 4352
  float* sV = smem + 8704;
  float* sA = smem + 13056;        // masked/scaled attention
  float* sU = smem + 17408;        // running state
  float* sG = smem + 21760;        // 64
  float* sR = smem + 21824;        // 64

  const int h = blockIdx.x;
  const int tid = threadIdx.x;
  const int lane = tid & 31;
  const int wave = tid >> 5;          // 0..15
  const int mi = (wave >> 2) * 16;    // tile row origin
  const int ni = (wave & 3) * 16;     // tile col origin
  const int n = lane & 15, half = lane >> 4;

  for (int i = tid; i < 64 * ALD; i += 512) sU[i] = 0.f;
  __syncthreads();

  for (int c = 0; c < L_SEQ / CHUNK; ++c) {
    const int t0 = c * CHUNK;
    const int gbase = t0 * DMODEL + h * DHEAD;

#if USE_TDM
    if (wave == 0) {
      tdm_load_tile(Qg + gbase, (unsigned)(unsigned long long)sQ);
      tdm_load_tile(Kg + gbase, (unsigned)(unsigned long long)sK);
      tdm_load_tile(Vg + gbase, (unsigned)(unsigned long long)sV);
      __builtin_amdgcn_s_wait_tensorcnt(0);
    }
#else
    for (int i = tid; i < 4096; i += 512) {
      const int row = i >> 6, col = i & 63;
      const int g = gbase + row * DMODEL + col;
      sQ[row * ALD + col] = Qg[g];
      sK[row * ALD + col] = Kg[g];
      sV[row * ALD + col] = Vg[g];
    }
#endif
    if (tid < CHUNK) {
      sG[tid] = Gc[(t0 + tid) * NHEAD + h];
      sR[tid] = rr[(t0 + tid) * NHEAD + h];
    }
    __syncthreads();

    // Phase 1: inter-chunk contribution  O = Qc @ U
    v8f acco = {};
    for (int k0 = 0; k0 < DHEAD; k0 += 4)
      acco = wmma_f32_k4(frag_a(sQ, ALD, mi, k0, lane),
                         frag_b(sU, ALD, k0, ni, lane), acco);

    // Phase 2: P = Qc @ Kc^T ; mask (s<=t) and scale by r[s]; A -> LDS.
    v8f accp = {};
    for (int k0 = 0; k0 < DHEAD; k0 += 4)
      accp = wmma_f32_k4(frag_a(sQ, ALD, mi, k0, lane),
                         frag_bt(sK, ALD, k0, ni, lane), accp);
    const int sl = ni + n;
    for (int v = 0; v < 8; ++v) {
      const int tl = mi + v + 8 * half;
      sA[tl * ALD + sl] = (sl <= tl) ? accp[v] * sR[sl] : 0.f;
    }
    __syncthreads();  // A complete; all waves past their U reads

    // Phase 3: O += A @ Vc ; emit rows scaled by Gc[t].
    for (int k0 = 0; k0 < CHUNK; k0 += 4)
      acco = wmma_f32_k4(frag_a(sA, ALD, mi, k0, lane),
                         frag_b(sV, ALD, k0, ni, lane), acco);
    for (int v = 0; v < 8; ++v) {
      const int tl = mi + v + 8 * half;
      Og[(t0 + tl) * DMODEL + h * DHEAD + ni + n] = acco[v] * sG[tl];
    }

    // Phase 4: U += (Kc * r)^T @ Vc   (tile (mi->d, ni->e), K-dim = s)
    v8f accu = {};
    for (int k0 = 0; k0 < CHUNK; k0 += 4) {
      const int s0 = k0 + 2 * half;
      const int d = mi + n;  // lane&15 indexes M (=d) of the A fragment
      v2f akr;
      akr.x = sK[s0 * ALD + d] * sR[s0];
      akr.y = sK[(s0 + 1) * ALD + d] * sR[s0 + 1];
      accu = wmma_f32_k4(akr, frag_b(sV, ALD, k0, ni, lane), accu);
    }
    for (int v = 0; v < 8; ++v) {
      const int drow = mi + v + 8 * half;
      sU[drow * ALD + ni + n] += accu[v];  // each element owned by one lane
    }
    __syncthreads();  // U + stage buffers quiesced before next chunk
  }
}

// ---------------------------------------------------------------------------
extern "C" void kernel_launch(void* const* d_in, const int* in_sizes, int n_in,
                              void* d_out, int out_size, void* d_ws,
                              size_t ws_size, hipStream_t stream) {
  const float* x  = (const float*)d_in[0];
  const float* Wq = (const float*)d_in[1];
  const float* Wk = (const float*)d_in[2];
  const float* Wv = (const float*)d_in[3];
  const float* Wo = (const float*)d_in[4];
  const float* gw = (const float*)d_in[5];
  const float* sw = (const float*)d_in[6];
  float* out = (float*)d_out;

  float* ws = (float*)d_ws;
  float* Q    = ws;                 // 1M floats
  float* K    = ws + 1048576;       // 1M
  float* V    = ws + 2097152;       // 1M
  float* attn = ws + 3145728;       // 1M
  float* logg = ws + 4194304;       // 16K
  float* Rv   = logg + 16384;       // 16K
  float* Gc   = Rv + 16384;         // 16K
  float* rr   = Gc + 16384;         // 16K   (total ~16.3 MB)

  const dim3 gg(DMODEL / 64, L_SEQ / 64);
  gemm_xwT_kernel<<<gg, 256, 0, stream>>>(x, Wq, Q);
  gemm_xwT_kernel<<<gg, 256, 0, stream>>>(x, Wk, K);
  gemm_xwT_kernel<<<gg, 256, 0, stream>>>(x, Wv, V);

  gate_kernel<<<(L_SEQ * NHEAD + 255) / 256, 256, 0, stream>>>(x, K, gw, sw,
                                                               logg, Rv);
  scan_kernel<<<NHEAD, 32, 0, stream>>>(logg, Rv, Gc, rr);

  gla_attn_kernel<<<NHEAD, 512, 21888 * sizeof(float), stream>>>(Q, K, V, Gc,
                                                                 rr, attn);

  gemm_xwT_kernel<<<gg, 256, 0, stream>>>(attn, Wo, out);
}